// Encoder_83425444758108
// MI455X (gfx1250) — compile-verified
//
#include <hip/hip_runtime.h>

// ---------------------------------------------------------------------------
// 2-layer GCN for MI455X (gfx1250, wave32).
//  - Dense h = X @ W via V_WMMA_F32_16X16X4_F32 (exact fp32 path).
//  - Edge scatter via hw fp32 global atomics (L2-resident working set).
// ---------------------------------------------------------------------------

typedef __attribute__((ext_vector_type(2))) float v2f;
typedef __attribute__((ext_vector_type(8))) float v8f;

// ---- degree / normalization ------------------------------------------------

__global__ void k_deg_init(int* __restrict__ deg, int n) {
    int i = blockIdx.x * blockDim.x + threadIdx.x;
    if (i < n) deg[i] = 1;              // self-loop
}

__global__ void k_deg_acc(const int* __restrict__ dst, int* __restrict__ deg, int E) {
    int e = blockIdx.x * blockDim.x + threadIdx.x;
    if (e < E) atomicAdd(&deg[dst[e]], 1);
}

__global__ void k_dinv(const int* __restrict__ deg, float* __restrict__ dinv, int n) {
    int i = blockIdx.x * blockDim.x + threadIdx.x;
    if (i < n) dinv[i] = rsqrtf((float)deg[i]);   // deg >= 1 always
}

// ---- GEMM: H[N,64] = X[N,64] @ W[64,64] via fp32 WMMA ----------------------

__global__ __launch_bounds__(256) void k_gemm64_wmma(
    const float* __restrict__ X, const float* __restrict__ W,
    float* __restrict__ H, int n) {
    const int lane  = threadIdx.x & 31;
    const int wave  = threadIdx.x >> 5;                 // 0..7
    const int gwave = blockIdx.x * 8 + wave;
    const int colTile   = gwave & 3;                    // 4 tiles of 16 cols
    int       rowTile   = gwave >> 2;
    const int rowTiles  = n >> 4;
    const int stride    = gridDim.x * 2;                // (gridDim*8)/4
    const int lr  = lane & 15;                          // m or n within tile
    const int hl  = lane >> 4;                          // 0|1 -> K half
    const int col0 = colTile * 16;

    // Preload this wave's B fragments: b[k] = W[4k+2*hl .. +1][col0+lr]
    v2f b[16];
#pragma unroll
    for (int k = 0; k < 16; ++k) {
        const int kr = k * 4 + hl * 2;
        b[k].x = W[kr * 64 + col0 + lr];
        b[k].y = W[(kr + 1) * 64 + col0 + lr];
    }

    for (; rowTile < rowTiles; rowTile += stride) {
        const int row0 = rowTile * 16;
        const float* xrow = X + (row0 + lr) * 64 + hl * 2;
        v8f acc = {};
#pragma unroll
        for (int k = 0; k < 16; ++k) {
            v2f a;
            a.x = xrow[k * 4];
            a.y = xrow[k * 4 + 1];
            acc = __builtin_amdgcn_wmma_f32_16x16x4_f32(
                /*neg_a=*/false, a, /*neg_b=*/false, b[k],
                /*c_mod=*/(short)0, acc, /*reuse_a=*/false, /*reuse_b=*/false);
        }
        float* hout = H + (row0 + hl * 8) * 64 + col0 + lr;
#pragma unroll
        for (int r = 0; r < 8; ++r) hout[r * 64] = acc[r];
    }
}

// Scalar tail for row counts not divisible by 16 (not hit for N=100000).
__global__ void k_gemm_tail(const float* __restrict__ X, const float* __restrict__ W,
                            float* __restrict__ H, int startRow, int n) {
    int idx = blockIdx.x * blockDim.x + threadIdx.x;
    int row = startRow + (idx >> 6);
    int col = idx & 63;
    if (row < n) {
        float acc = 0.f;
        for (int k = 0; k < 64; ++k) acc += X[row * 64 + k] * W[k * 64 + col];
        H[row * 64 + col] = acc;
    }
}

// ---- out = bias + dinv[i]^2 * h[i]  (self-loop term, full overwrite) -------

__global__ void k_node_init(float* __restrict__ out, const float* __restrict__ h,
                            const float* __restrict__ dinv,
                            const float* __restrict__ bias, int n) {
    int i = blockIdx.x * blockDim.x + threadIdx.x;
    if (i < n * 64) {
        int node = i >> 6, c = i & 63;
        float di = dinv[node];
        out[i] = bias[c] + h[i] * di * di;
    }
}

// ---- edge scatter: one wave per edge, 2 floats per lane --------------------

__global__ void k_edge_scatter(const int* __restrict__ src, const int* __restrict__ dst,
                               const float* __restrict__ dinv,
                               const float* __restrict__ h,
                               float* __restrict__ out, int E) {
    int wv   = (int)((blockIdx.x * (unsigned)blockDim.x + threadIdx.x) >> 5);
    int lane = threadIdx.x & 31;
    if (wv >= E) return;
    int s = src[wv], d = dst[wv];
    float norm = dinv[s] * dinv[d];
    const float2* hp = (const float2*)(h + s * 64);
    float2 v = hp[lane];
    float* op = out + d * 64 + lane * 2;
    unsafeAtomicAdd(op,     v.x * norm);
    unsafeAtomicAdd(op + 1, v.y * norm);
}

// ---- ReLU ------------------------------------------------------------------

__global__ void k_relu(float* __restrict__ x, int n) {
    int i = blockIdx.x * blockDim.x + threadIdx.x;
    if (i < n) x[i] = fmaxf(x[i], 0.0f);
}

// ---------------------------------------------------------------------------

extern "C" void kernel_launch(void* const* d_in, const int* in_sizes, int n_in,
                              void* d_out, int out_size, void* d_ws, size_t ws_size,
                              hipStream_t stream) {
    const float* x  = (const float*)d_in[0];
    const int*   ei = (const int*)d_in[1];
    const float* W1 = (const float*)d_in[2];
    const float* b1 = (const float*)d_in[3];
    const float* W2 = (const float*)d_in[4];
    const float* b2 = (const float*)d_in[5];

    const int N = in_sizes[0] / 64;
    const int E = in_sizes[1] / 2;
    const int* src = ei;
    const int* dst = ei + E;

    // Workspace layout: deg[N] (int) | dinv[N] (f32) | hbuf[N*64] (f32)
    int*   deg  = (int*)d_ws;
    float* dinv = (float*)d_ws + N;
    float* hbuf = (float*)d_ws + 2 * (size_t)N;
    float* out  = (float*)d_out;

    const int T = 256;
    const int nb_n  = (N + T - 1) / T;
    const int nb_e  = (E + T - 1) / T;
    const int nb_nf = (N * 64 + T - 1) / T;
    const int nb_sc = (int)(((long long)E * 32 + T - 1) / T);

    // ---- normalization (shared by both layers) ----
    k_deg_init<<<nb_n, T, 0, stream>>>(deg, N);
    k_deg_acc <<<nb_e, T, 0, stream>>>(dst, deg, E);
    k_dinv    <<<nb_n, T, 0, stream>>>(deg, dinv, N);

    const int rowTiles = N >> 4;
    const int gb = (rowTiles * 4 + 7) / 8;
    const int tailStart = rowTiles * 16;
    const int tailRows  = N - tailStart;

    // ---- layer 1: h1 = x@W1 ; out = b1 + dinv^2*h1 ; scatter ; relu ----
    if (rowTiles > 0)
        k_gemm64_wmma<<<gb, T, 0, stream>>>(x, W1, hbuf, N);
    if (tailRows > 0)
        k_gemm_tail<<<(tailRows * 64 + T - 1) / T, T, 0, stream>>>(x, W1, hbuf, tailStart, N);
    k_node_init   <<<nb_nf, T, 0, stream>>>(out, hbuf, dinv, b1, N);
    k_edge_scatter<<<nb_sc, T, 0, stream>>>(src, dst, dinv, hbuf, out, E);
    k_relu        <<<nb_nf, T, 0, stream>>>(out, N * 64);

    // ---- layer 2: h2 = out@W2 ; out = b2 + dinv^2*h2 ; scatter ; relu ----
    if (rowTiles > 0)
        k_gemm64_wmma<<<gb, T, 0, stream>>>(out, W2, hbuf, N);
    if (tailRows > 0)
        k_gemm_tail<<<(tailRows * 64 + T - 1) / T, T, 0, stream>>>(out, W2, hbuf, tailStart, N);
    k_node_init   <<<nb_nf, T, 0, stream>>>(out, hbuf, dinv, b2, N);
    k_edge_scatter<<<nb_sc, T, 0, stream>>>(src, dst, dinv, hbuf, out, E);
    k_relu        <<<nb_nf, T, 0, stream>>>(out, N * 64);
}